// NodeToEdge_50560355008916
// MI455X (gfx1250) — compile-verified
//
#include <hip/hip_runtime.h>

// NodeToEdge gather-multiply for MI455X (gfx1250, wave32).
//
// Bandwidth-bound (~0.15 FLOP/byte -> HBM roofline ~66us @ 23.3 TB/s).
// Data-movement plan:
//  - node tables (2 x 51.2 MB) fit the 192 MB L2 -> regular-temporal loads
//  - off_edge streams + output are touch-once -> non-temporal (th:TH_*_NT)
//    so they never evict the node tables
//  - one edge row per wave32: edge index is exactly wave-uniform ->
//    readfirstlane + s_load_b32 for the two ids (KMcnt path, zero VMEM)
//  - each thread moves 2 float4s (slots r and r+32: 512-byte immediate
//    offset, clause-friendly b128 pairs); all traffic is b128
//  - 64-bit row offsets (row index cast to long long before * FEAT)

#define FEAT 256
#define ROW_VECS 64            // float4 slots per edge row
#define BLOCK 256              // 8 wave32s -> 8 edge rows per block
#define ROWS_PER_BLOCK 8

typedef float v4f __attribute__((ext_vector_type(4)));

__global__ __launch_bounds__(BLOCK) void NodeToEdge_50560355008916_kernel(
    const float* __restrict__ node_src,   // [N, 256]
    const float* __restrict__ node_tgt,   // [N, 256]
    const int*   __restrict__ edge_ids,   // [2, E] (int32)
    const float* __restrict__ off_src,    // [E, 256]
    const float* __restrict__ off_tgt,    // [E, 256]
    float*       __restrict__ out,        // [E, 256]
    int num_edges)
{
    const int tid = threadIdx.x;
    int e = blockIdx.x * ROWS_PER_BLOCK + (tid >> 5);  // one row per wave32
    const int r = tid & 31;                            // float4 slot 0..31
    if (e >= num_edges) return;

    // e is wave-uniform by construction (tid>>5 == wave id); prove it so the
    // id fetches become scalar loads.
    e = __builtin_amdgcn_readfirstlane(e);
    const int src_id = edge_ids[e];               // row 0 of [2,E]
    const int tgt_id = edge_ids[num_edges + e];   // row 1 of [2,E]

    const v4f* ns = (const v4f*)(node_src + (long long)src_id * FEAT) + r;
    const v4f* nt = (const v4f*)(node_tgt + (long long)tgt_id * FEAT) + r;
    const v4f* os = (const v4f*)(off_src  + (long long)e * FEAT) + r;
    const v4f* ot = (const v4f*)(off_tgt  + (long long)e * FEAT) + r;
    v4f*       op = (v4f*)      (out      + (long long)e * FEAT) + r;

    // Streaming operands: non-temporal (touch-once).
    v4f c0 = __builtin_nontemporal_load(os);
    v4f c1 = __builtin_nontemporal_load(os + 32);   // +512B imm offset
    v4f d0 = __builtin_nontemporal_load(ot);
    v4f d1 = __builtin_nontemporal_load(ot + 32);

    // Gathered node rows: regular-temporal (L2-resident tables).
    v4f a0 = ns[0];
    v4f a1 = ns[32];
    v4f b0 = nt[0];
    v4f b1 = nt[32];

    v4f r0 = (a0 + c0) * (b0 + d0);
    v4f r1 = (a1 + c1) * (b1 + d1);

    // Write-once output: non-temporal stores.
    __builtin_nontemporal_store(r0, op);
    __builtin_nontemporal_store(r1, op + 32);
}

extern "C" void kernel_launch(void* const* d_in, const int* in_sizes, int n_in,
                              void* d_out, int out_size, void* d_ws, size_t ws_size,
                              hipStream_t stream) {
    const float* node_src = (const float*)d_in[0];
    const float* node_tgt = (const float*)d_in[1];
    const int*   edge_ids = (const int*)d_in[2];
    const float* off_src  = (const float*)d_in[3];
    const float* off_tgt  = (const float*)d_in[4];
    float*       out      = (float*)d_out;

    const int num_edges = in_sizes[3] / FEAT;   // off_edge_src is [E, 256]

    const unsigned blocks =
        (unsigned)((num_edges + ROWS_PER_BLOCK - 1) / ROWS_PER_BLOCK);

    NodeToEdge_50560355008916_kernel<<<blocks, BLOCK, 0, stream>>>(
        node_src, node_tgt, edge_ids, off_src, off_tgt, out, num_edges);
}